// EMACodebook_91216515432660
// MI455X (gfx1250) — compile-verified
//
#include <hip/hip_runtime.h>
#include <stdint.h>
#include <stddef.h>

// ---------------------------------------------------------------------------
// EMACodebook.update_codebook for MI455X (gfx1250), compile-only tuning.
//
// Shapes (from reference): BN=524288, D=256, K=1024.
// Memory-roofline problem: ~1.03 GB total traffic -> ~45us at 23.3 TB/s.
// WMMA rejected by arithmetic (one-hot GEMM = 275 GFLOP vs 0.67 GFLOP needed).
// CDNA5 paths used: TDM tensor_load_to_lds (TENSORcnt) for tile staging,
// ds_add_f32 LDS-privatized scatter, global_atomic_add_f32 flush.
// ---------------------------------------------------------------------------

#define D_DIM 256
#define K_CLASSES 1024
#define SLICE 64                         // columns per slice
#define NSLICE (D_DIM / SLICE)           // 4
#define TILE_ROWS 64
#define NWG 64                           // workgroups per slice in scatter pass
#define WG_THREADS 256

#define ACC_FLOATS (K_CLASSES * SLICE)   // 65536 floats = 256 KB
#define STAGE_FLOATS (TILE_ROWS * SLICE) // 4096 floats  = 16 KB
#define LDS_TOTAL_BYTES ((ACC_FLOATS + 2 * STAGE_FLOATS) * 4)  // 294912 < 320KB

typedef float f32x4 __attribute__((ext_vector_type(4)));
typedef unsigned int u32x4 __attribute__((ext_vector_type(4)));
typedef unsigned int u32x8 __attribute__((ext_vector_type(8)));

#define MOMENTUM 0.9f
#define EPS_NORM 1e-12f
#define EPS_CNT  1e-6f

// --------------------------- kernel 0: zero workspace ----------------------
__global__ void ema_init_ws(float* __restrict__ sums, unsigned* __restrict__ cnt) {
  int i = blockIdx.x * blockDim.x + threadIdx.x;
  if (i < K_CLASSES * D_DIM) sums[i] = 0.0f;
  if (i < K_CLASSES) cnt[i] = 0u;
}

// ------------------- kernel 1: row inverse norms + class counts ------------
// Wave-per-row: 32 lanes x 8 floats = 256 columns. Nontemporal b128 streams.
__global__ void ema_rownorm(const float* __restrict__ patch,
                            const int* __restrict__ labels2,   // int64 viewed as int pairs
                            float* __restrict__ inv_norm,
                            unsigned* __restrict__ cnt,
                            int BN) {
  int row  = (int)((blockIdx.x * (unsigned)blockDim.x + threadIdx.x) >> 5);
  int lane = threadIdx.x & 31;
  if (row >= BN) return;

  const f32x4* p = (const f32x4*)(patch + (size_t)row * D_DIM) + lane * 2;
  f32x4 a = __builtin_nontemporal_load(p);
  f32x4 b = __builtin_nontemporal_load(p + 1);
  float ss = a.x * a.x + a.y * a.y + a.z * a.z + a.w * a.w
           + b.x * b.x + b.y * b.y + b.z * b.z + b.w * b.w;
#pragma unroll
  for (int m = 16; m >= 1; m >>= 1) ss += __shfl_xor(ss, m, 32);

  float inv = 1.0f / fmaxf(sqrtf(ss), EPS_NORM);
  if (lane == 0) {
    inv_norm[row] = inv;
    atomicAdd(&cnt[labels2[2 * (size_t)row]], 1u);   // low dword of int64 label
  }
}

// ------------------- kernel 2: TDM-staged LDS-privatized scatter -----------
// grid = (NWG, NSLICE). LDS: [1024 x 64] f32 accumulator + 2 staging tiles.
// One TDM descriptor per 64x64 tile (16KB), double-buffered on TENSORcnt.
__global__ void ema_scatter(const float* __restrict__ patch,
                            const int* __restrict__ labels2,
                            const float* __restrict__ inv_norm,
                            float* __restrict__ sums,
                            int BN) {
  extern __shared__ float lds[];  // offset 0 of the workgroup's LDS allocation

  const int tid = threadIdx.x;
  const int sli = blockIdx.y;                 // column slice 0..3
  const int wg  = blockIdx.x;
  const int rowsPerWG = BN / NWG;             // 8192
  const int numTiles  = rowsPerWG / TILE_ROWS;
  const int startRow  = wg * rowsPerWG;
  const uint32_t colBase = (uint32_t)sli * SLICE;

  for (int i = tid; i < ACC_FLOATS; i += WG_THREADS) lds[i] = 0.0f;

  const uint32_t stageByte0 = (uint32_t)ACC_FLOATS * 4u;
  const uint32_t stageByte1 = stageByte0 + (uint32_t)STAGE_FLOATS * 4u;

  // Issue one 64-row x 64-col f32 tile via the Tensor Data Mover.
  // D# group 0: {count=1 | lds_addr | global_addr | type=2}; group 1 constant:
  // data_size=4B, tensor_dim0=64, tensor_dim1=64, tile_dim0=64, tile_dim1=64,
  // tensor_dim0_stride=256 elements (full row pitch). Tile == tensor -> no OOB.
  auto issue = [&](int tile, uint32_t stageByte) {
    uint64_t gaddr = (uint64_t)(uintptr_t)patch +
        ((uint64_t)((uint32_t)(startRow + tile * TILE_ROWS)) * (uint32_t)D_DIM
         + colBase) * 4u;
    u32x4 g0 = { 1u,                                   // count=1, user mode
                 stageByte,                            // lds_addr (bytes)
                 (uint32_t)gaddr,                      // global_addr[31:0]
                 ((uint32_t)(gaddr >> 32) & 0x01FFFFFFu) | 0x80000000u }; // [56:32] | type=2
    u32x8 g1 = { 0x00020000u,                          // data_size=2 (4 bytes)
                 (uint32_t)TILE_ROWS << 16,            // tensor_dim0[15:0]=64
                 (uint32_t)TILE_ROWS << 16,            // tensor_dim0 hi=0 | tensor_dim1[15:0]=64
                 (uint32_t)SLICE << 16,                // tensor_dim1 hi=0 | tile_dim0=64
                 (uint32_t)TILE_ROWS,                  // tile_dim1=64 | tile_dim2=0
                 (uint32_t)D_DIM,                      // tensor_dim0_stride=256 elements
                 0u, 0u };                             // stride hi / dim1_stride=0
    asm volatile("tensor_load_to_lds %0, %1" :: "s"(g0), "s"(g1) : "memory");
  };

  if (tid < 32) issue(0, stageByte0);   // wave 0 drives the TDM

  const int g = tid >> 6;   // row group 0..3 (wave pairs share a row)
  const int c = tid & 63;   // column within slice

  for (int t = 0; t < numTiles; ++t) {
    const int buf = t & 1;
    if (tid < 32) {
      if (t + 1 < numTiles) {
        issue(t + 1, buf ? stageByte0 : stageByte1);
        __builtin_amdgcn_s_wait_tensorcnt(1);   // tile t landed, t+1 in flight
      } else {
        __builtin_amdgcn_s_wait_tensorcnt(0);
      }
    }
    __syncthreads();   // publish tile t's LDS to all 8 waves (also covers init)

    const float* stg = lds + ACC_FLOATS + buf * STAGE_FLOATS;
    const int tileRow0 = startRow + t * TILE_ROWS;
#pragma unroll 4
    for (int r = g; r < TILE_ROWS; r += 4) {
      const int row = tileRow0 + r;
      const float v = stg[r * SLICE + c] * inv_norm[row];
      const int lab = labels2[2 * (size_t)row];
      atomicAdd(&lds[lab * SLICE + c], v);   // ds_add_f32, conflict-free banks
    }
    __syncthreads();  // all consumed before this buffer is overwritten next iter
  }

  // Flush the privatized accumulator: 64K device-scope f32 atomics per WG.
  for (int i = tid; i < ACC_FLOATS; i += WG_THREADS) {
    const float v = lds[i];
    if (v != 0.0f) {
      uint32_t kcls = (uint32_t)i >> 6;
      uint32_t cc   = (uint32_t)i & 63u;
      uint32_t voff = (kcls * (uint32_t)D_DIM + colBase + cc) * 4u;
      asm volatile("global_atomic_add_f32 %0, %1, %2 scope:SCOPE_DEV"
                   :: "v"(voff), "v"(v), "s"(sums)
                   : "memory");
    }
  }
}

// ------------------- kernel 3: finalize (EMA + renorm + select) ------------
// Wave-per-class: 32 lanes x 8 cols.
__global__ void ema_finalize(const float* __restrict__ sums,
                             const unsigned* __restrict__ cnt,
                             const float* __restrict__ codebook,
                             const int* __restrict__ counts,
                             float* __restrict__ out_cb,
                             int* __restrict__ out_cnt) {
  int kcls = (int)((blockIdx.x * (unsigned)blockDim.x + threadIdx.x) >> 5);
  int lane = threadIdx.x & 31;
  if (kcls >= K_CLASSES) return;

  const float* srow = sums + (size_t)kcls * D_DIM + lane * 8;
  const float* crow = codebook + (size_t)kcls * D_DIM + lane * 8;

  const unsigned n = cnt[kcls];
  const float denom = fmaxf((float)n, EPS_CNT);

  float p[8], cb[8], e[8];
  float ss = 0.0f;
#pragma unroll
  for (int i = 0; i < 8; ++i) {
    p[i]  = srow[i] / denom;
    cb[i] = crow[i];
    ss += p[i] * p[i];
  }
#pragma unroll
  for (int m = 16; m >= 1; m >>= 1) ss += __shfl_xor(ss, m, 32);
  const float inv1 = 1.0f / fmaxf(sqrtf(ss), EPS_NORM);

  float ss2 = 0.0f;
#pragma unroll
  for (int i = 0; i < 8; ++i) {
    p[i] *= inv1;                                  // class_proto
    e[i] = MOMENTUM * cb[i] + (1.0f - MOMENTUM) * p[i];
    ss2 += e[i] * e[i];
  }
#pragma unroll
  for (int m = 16; m >= 1; m >>= 1) ss2 += __shfl_xor(ss2, m, 32);
  const float inv2 = 1.0f / fmaxf(sqrtf(ss2), EPS_NORM);

  const bool present = n > 0u;
  const int oldc = counts[kcls];
#pragma unroll
  for (int i = 0; i < 8; ++i) {
    const float ema = e[i] * inv2;
    const float nv  = (oldc == 0) ? p[i] : ema;
    out_cb[(size_t)kcls * D_DIM + lane * 8 + i] = present ? nv : cb[i];
  }
  if (lane == 0) out_cnt[kcls] = oldc + (present ? 1 : 0);
}

// ---------------------------------------------------------------------------
extern "C" void kernel_launch(void* const* d_in, const int* in_sizes, int n_in,
                              void* d_out, int out_size, void* d_ws, size_t ws_size,
                              hipStream_t stream) {
  const float* patch    = (const float*)d_in[0];
  const int*   labels2  = (const int*)d_in[1];   // int64 labels, little-endian pairs
  const float* codebook = (const float*)d_in[2];
  const int*   counts   = (const int*)d_in[3];

  const int BN = in_sizes[1];                    // 524288

  float*    out_cb  = (float*)d_out;
  int*      out_cnt = (int*)(out_cb + (size_t)K_CLASSES * D_DIM);

  // workspace layout: sums [K*D] f32 | cnt [K] u32 | inv_norm [BN] f32  (~3.1 MB)
  float*    sums     = (float*)d_ws;
  unsigned* cnt      = (unsigned*)(sums + (size_t)K_CLASSES * D_DIM);
  float*    inv_norm = (float*)(cnt + K_CLASSES);

  // Allow the 288KB dynamic-LDS launch (gfx1250 WGP = 320KB). Host-side,
  // idempotent, not a stream op -> graph-capture safe.
  hipFuncSetAttribute((const void*)ema_scatter,
                      hipFuncAttributeMaxDynamicSharedMemorySize, LDS_TOTAL_BYTES);

  // 0) zero accumulators
  {
    int total = K_CLASSES * D_DIM;
    ema_init_ws<<<(total + WG_THREADS - 1) / WG_THREADS, WG_THREADS, 0, stream>>>(sums, cnt);
  }
  // 1) row norms + class counts (wave-per-row)
  {
    int blocks = (BN * 32 + WG_THREADS - 1) / WG_THREADS;
    ema_rownorm<<<blocks, WG_THREADS, 0, stream>>>(patch, labels2, inv_norm, cnt, BN);
  }
  // 2) TDM-staged, LDS-privatized scatter into per-class sums
  {
    dim3 grid(NWG, NSLICE, 1);
    ema_scatter<<<grid, WG_THREADS, LDS_TOTAL_BYTES, stream>>>(patch, labels2, inv_norm, sums, BN);
  }
  // 3) finalize: EMA + renormalize + select, write codebook and counts
  {
    int blocks = (K_CLASSES * 32 + WG_THREADS - 1) / WG_THREADS;
    ema_finalize<<<blocks, WG_THREADS, 0, stream>>>(sums, cnt, codebook, counts, out_cb, out_cnt);
  }
}